// Rep_Separate_SelfAttention_28930899706120
// MI455X (gfx1250) — compile-verified
//
#include <hip/hip_runtime.h>
#include <hip/hip_bf16.h>

// ---------------------------------------------------------------------------
// Rep_Separate_SelfAttention for MI455X (gfx1250, wave32, WMMA).
//
// Bandwidth-bound on the mandatory 543MB attn output (>=1.1GB traffic ~47us
// @ 23.3TB/s), so all GEMMs stay fp32 via V_WMMA_F32_16X16X4_F32 (no need to
// trade precision for flops). Scores+softmax fused with a 16x1040 fp32 LDS
// strip (~69KB, fine under CDNA5's 320KB/WGP). All tile staging uses b128/b64
// vector moves; the GEMM B-tile additionally uses the Tensor Data Mover
// (tensor_load_to_lds + s_wait_tensorcnt) with TDM LDS-padding reproducing
// the +4-DWORD bank-conflict padding. Workspace: 5*8240*1024 f32 (~169 MB).
// ---------------------------------------------------------------------------

#define BB   8
#define LL   1030
#define DD   1024
#define HH   16
#define DKV  64
#define NTOK (BB * LL)        // 8240
#define LT   65               // ceil(L/16)
#define IN_SN 1027            // tokens >= IN_SN use weight set 1

typedef float v2f __attribute__((ext_vector_type(2)));
typedef float v4f __attribute__((ext_vector_type(4)));
typedef float v8f __attribute__((ext_vector_type(8)));
typedef unsigned int u32x4 __attribute__((ext_vector_type(4)));
typedef int i32x4 __attribute__((ext_vector_type(4)));
typedef int i32x8 __attribute__((ext_vector_type(8)));

#if defined(__has_builtin)
#if __has_builtin(__builtin_amdgcn_tensor_load_to_lds) && \
    __has_builtin(__builtin_amdgcn_s_wait_tensorcnt)
#define HAVE_TDM 1
#endif
#endif
#ifndef HAVE_TDM
#define HAVE_TDM 0
#endif

__device__ __forceinline__ v8f wmma_f32(v2f a, v2f b, v8f c) {
  // D = A(16x4, f32) * B(4x16, f32) + C(16x16, f32)
  return __builtin_amdgcn_wmma_f32_16x16x4_f32(
      /*neg_a=*/false, a, /*neg_b=*/false, b,
      /*c_mod=*/(short)0, c, /*reuse_a=*/false, /*reuse_b=*/false);
}

#if HAVE_TDM
// TDM: load a 16x64 f32 tile (row stride 1024 elems) from gsrc into LDS at
// lds_off, inserting 4 DWORDs of padding every 64 DWORDs (-> Bs[16][68]).
// D# bit layout per CDNA5 ISA.
__device__ __forceinline__ void tdm_load_tile16x64(const float* gsrc,
                                                   unsigned lds_off) {
  unsigned long long ga = (unsigned long long)(size_t)gsrc;
  u32x4 g0;
  g0.x = 1u;                                             // count=1 (valid D#)
  g0.y = lds_off;                                        // lds_addr
  g0.z = (unsigned)(ga & 0xffffffffu);                   // global_addr lo
  g0.w = (unsigned)((ga >> 32) & 0x01ffffffu) | (2u << 30);  // addr hi | type=2

  unsigned long long q0 = (2ull << 16)                   // data_size = 4B
                        | (1ull << 20)                   // pad_enable
                        | (5ull << 22)                   // pad_interval: 64 DW
                        | (3ull << 25)                   // pad_amount: 4 DW
                        | ((unsigned long long)DD << 48);// tensor_dim0 (lo 16)
  unsigned long long q1 = ((unsigned long long)DD << 16) // tensor_dim1
                        | (64ull << 48);                 // tile_dim0 = 64
  unsigned long long q2 = 16ull                          // tile_dim1 = 16
                        | ((unsigned long long)DD << 32);// tensor_dim0_stride
  unsigned long long q3 = ((unsigned long long)DD << 16);// tensor_dim1_stride
  i32x8 g1;
  g1[0] = (int)(q0 & 0xffffffffu); g1[1] = (int)(q0 >> 32);
  g1[2] = (int)(q1 & 0xffffffffu); g1[3] = (int)(q1 >> 32);
  g1[4] = (int)(q2 & 0xffffffffu); g1[5] = (int)(q2 >> 32);
  g1[6] = (int)(q3 & 0xffffffffu); g1[7] = (int)(q3 >> 32);
  i32x4 gz = {0, 0, 0, 0};
#if __clang_major__ >= 23
  i32x8 gz8 = {0, 0, 0, 0, 0, 0, 0, 0};
  __builtin_amdgcn_tensor_load_to_lds(g0, g1, gz, gz, gz8, 0);
#else
  __builtin_amdgcn_tensor_load_to_lds(g0, g1, gz, gz, 0);
#endif
}
#endif

// ---------------------------------------------------------------------------
// 1) LayerNorm: one block per token, D=1024, 256 threads x float4.
// ---------------------------------------------------------------------------
__global__ __launch_bounds__(256) void ln_kernel(
    const float* __restrict__ x, const float* __restrict__ g,
    const float* __restrict__ beta, float* __restrict__ xn) {
  __shared__ float red[256];
  const int tid = threadIdx.x;
  const float* xr = x + (size_t)blockIdx.x * DD;
  const v4f vx = *(const v4f*)(xr + tid * 4);
  float s = vx.x + vx.y + vx.z + vx.w;
  red[tid] = s;
  __syncthreads();
  for (int off = 128; off > 0; off >>= 1) {
    if (tid < off) red[tid] += red[tid + off];
    __syncthreads();
  }
  const float mu = red[0] * (1.f / (float)DD);
  __syncthreads();
  v4f d = vx - mu;
  red[tid] = d.x * d.x + d.y * d.y + d.z * d.z + d.w * d.w;
  __syncthreads();
  for (int off = 128; off > 0; off >>= 1) {
    if (tid < off) red[tid] += red[tid + off];
    __syncthreads();
  }
  const float rstd = rsqrtf(red[0] * (1.f / (float)DD) + 1e-6f);
  const v4f vg = *(const v4f*)(g + tid * 4);
  const v4f vb = *(const v4f*)(beta + tid * 4);
  v4f o = d * rstd * vg + vb;
  *(v4f*)(xn + (size_t)blockIdx.x * DD + tid * 4) = o;
}

// ---------------------------------------------------------------------------
// 2) fp32 WMMA GEMM: C[M,N] = A[M,K] * W[K,N] (+bias +residual).
//    64x64 block tile, TK=16, 8 waves (2x4), 2 accumulators per wave.
//    A staged via b128 loads; B staged via TDM (wave 0) when available.
// ---------------------------------------------------------------------------
__global__ __launch_bounds__(256) void gemm64_kernel(
    const float* __restrict__ A, const float* __restrict__ W,
    float* __restrict__ C, int M, int K, int N,
    const float* __restrict__ bias, const float* __restrict__ resid) {
  __shared__ float As[64][20];   // +4 DW pad -> conflict-free frag reads
  __shared__ float Bs[16][68];   // +4 DW pad (TDM pad_interval/amount match)
  const int tid = threadIdx.x;
  const int wid = tid >> 5, lane = tid & 31;
  const int wm = wid >> 2, wn = wid & 3;
  const int m0 = blockIdx.y * 64, n0 = blockIdx.x * 64;
  const int lrow = lane & 15, kb = (lane >> 4) * 2;

  // A tile: thread -> row tid>>2, 4 cols (tid&3)*4
  const int ar = tid >> 2, ac = (tid & 3) * 4;
  const int agr = m0 + ar;
  // B tile fallback: thread -> row tid>>4, 4 cols (tid&15)*4
  const int br = tid >> 4, bc = (tid & 15) * 4;

  v8f acc0 = {0.f, 0.f, 0.f, 0.f, 0.f, 0.f, 0.f, 0.f};
  v8f acc1 = acc0;

  for (int kt = 0; kt < K; kt += 16) {
#if HAVE_TDM
    if (wid == 0)
      tdm_load_tile16x64(W + (size_t)kt * N + n0,
                         (unsigned)(size_t)(void*)&Bs[0][0]);
#else
    *(v4f*)&Bs[br][bc] = *(const v4f*)(W + (size_t)(kt + br) * N + n0 + bc);
#endif
    v4f av = {0.f, 0.f, 0.f, 0.f};
    if (agr < M) av = *(const v4f*)(A + (size_t)agr * K + kt + ac);
    *(v4f*)&As[ar][ac] = av;
    if (kt + 16 < K && agr < M)   // prefetch next A tile -> global_prefetch_b8
      __builtin_prefetch(A + (size_t)agr * K + kt + 16 + ac, 0, 1);
#if HAVE_TDM
    if (wid == 0) __builtin_amdgcn_s_wait_tensorcnt(0);
#endif
    __syncthreads();
#pragma unroll
    for (int ks = 0; ks < 16; ks += 4) {
      v2f b;
      b.x = Bs[ks + kb][wn * 16 + lrow];
      b.y = Bs[ks + kb + 1][wn * 16 + lrow];
      v2f a0, a1;
      a0.x = As[wm * 32 + lrow][ks + kb];
      a0.y = As[wm * 32 + lrow][ks + kb + 1];
      a1.x = As[wm * 32 + 16 + lrow][ks + kb];
      a1.y = As[wm * 32 + 16 + lrow][ks + kb + 1];
      acc0 = wmma_f32(a0, b, acc0);
      acc1 = wmma_f32(a1, b, acc1);
    }
    __syncthreads();
  }

  const int col = n0 + wn * 16 + lrow;
  const int rh = (lane >> 4) * 8;   // C/D: VGPR i <-> rows M=i and M=8+i
  const float bv = bias ? bias[col] : 0.f;
#pragma unroll
  for (int i = 0; i < 8; ++i) {
    int gr0 = m0 + wm * 32 + i + rh;
    if (gr0 < M) {
      float v = acc0[i] + bv;
      if (resid) v += resid[(size_t)gr0 * N + col];
      C[(size_t)gr0 * N + col] = v;
    }
    int gr1 = gr0 + 16;
    if (gr1 < M) {
      float v = acc1[i] + bv;
      if (resid) v += resid[(size_t)gr1 * N + col];
      C[(size_t)gr1 * N + col] = v;
    }
  }
}

// ---------------------------------------------------------------------------
// 3) Fixup: recompute the 3 rep tokens/batch with weight set 1 (q/k/v).
// ---------------------------------------------------------------------------
__global__ __launch_bounds__(256) void fixup_kernel(
    const float* __restrict__ xn,
    const float* __restrict__ wq1, const float* __restrict__ wk1,
    const float* __restrict__ wv1,
    float* __restrict__ q, float* __restrict__ k, float* __restrict__ v) {
  const int which = blockIdx.y;
  const float* w = (which == 0) ? wq1 : (which == 1) ? wk1 : wv1;
  float* dst = (which == 0) ? q : (which == 1) ? k : v;
  const int b = blockIdx.x / 3, s = blockIdx.x % 3;
  const int row = b * LL + IN_SN + s;
  const float* xr = xn + (size_t)row * DD;
  float acc[4] = {0.f, 0.f, 0.f, 0.f};
  for (int kk = 0; kk < DD; ++kk) {
    float xv = xr[kk];
#pragma unroll
    for (int j = 0; j < 4; ++j)
      acc[j] += xv * w[(size_t)kk * DD + threadIdx.x + j * 256];
  }
#pragma unroll
  for (int j = 0; j < 4; ++j)
    dst[(size_t)row * DD + threadIdx.x + j * 256] = acc[j];
}

// ---------------------------------------------------------------------------
// 4) Fused scores + softmax per (bh, 16-row q-tile).
//    16x1040 fp32 score strip in dynamic LDS; only final probs hit HBM.
// ---------------------------------------------------------------------------
__global__ __launch_bounds__(256) void scores_softmax_kernel(
    const float* __restrict__ qm, const float* __restrict__ km,
    float* __restrict__ attn) {
  extern __shared__ float smem[];
  float* S  = smem;               // [16][1040]
  float* Qs = smem + 16 * 1040;   // [16][68]
  const int qt = blockIdx.x;      // 0..64
  const int bh = blockIdx.y;      // 0..127
  const int b = bh >> 4, h = bh & 15;
  const float* qb = qm + (size_t)b * LL * DD + h * DKV;
  const float* kbp = km + (size_t)b * LL * DD + h * DKV;
  const int tid = threadIdx.x, wid = tid >> 5, lane = tid & 31;
  const int lrow = lane & 15, kb = (lane >> 4) * 2;

  {  // Q tile 16x64: thread -> row tid>>4, 4 cols (tid&15)*4 (b128 moves)
    const int r = tid >> 4, c4 = (tid & 15) * 4;
    const int ql = qt * 16 + r;
    v4f qv = {0.f, 0.f, 0.f, 0.f};
    if (ql < LL) qv = *(const v4f*)(qb + (size_t)ql * DD + c4);
    *(v4f*)&Qs[r * 68 + c4] = qv;
  }
  __syncthreads();

  for (int kt = wid; kt < LT; kt += 8) {   // wave owns k-tiles wid, wid+8, ...
    v8f acc = {0.f, 0.f, 0.f, 0.f, 0.f, 0.f, 0.f, 0.f};
    const int key = kt * 16 + lrow;
    const bool kvalid = key < LL;
    const float2* kr2 = (const float2*)(kbp + (size_t)key * DD);
#pragma unroll
    for (int ks = 0; ks < DKV; ks += 4) {
      v2f a, bf;
      a.x = Qs[lrow * 68 + ks + kb];
      a.y = Qs[lrow * 68 + ks + kb + 1];
      if (kvalid) {                        // B = K^T gather, one b64 per step
        float2 kv = kr2[(ks + kb) >> 1];
        bf.x = kv.x; bf.y = kv.y;
      } else { bf.x = 0.f; bf.y = 0.f; }
      acc = wmma_f32(a, bf, acc);
    }
    const int rh = (lane >> 4) * 8;
#pragma unroll
    for (int i = 0; i < 8; ++i)
      S[(i + rh) * 1040 + kt * 16 + lrow] = acc[i] * 0.125f;  // 1/sqrt(64)
  }
  __syncthreads();

  // softmax over rows (wave per row, 2 rows per wave); LL = 2*515 -> float2
  for (int r = wid; r < 16; r += 8) {
    const int qrow = qt * 16 + r;
    if (qrow >= LL) continue;
    float2* Sr = (float2*)&S[r * 1040];
    float m = -3.4e38f;
    for (int j = lane; j < 515; j += 32) {
      float2 sv = Sr[j];
      m = fmaxf(m, fmaxf(sv.x, sv.y));
    }
#pragma unroll
    for (int off = 16; off > 0; off >>= 1) m = fmaxf(m, __shfl_xor(m, off, 32));
    float s = 0.f;
    for (int j = lane; j < 515; j += 32) {
      float2 sv = Sr[j];
      float2 e = make_float2(__expf(sv.x - m), __expf(sv.y - m));
      Sr[j] = e;
      s += e.x + e.y;
    }
#pragma unroll
    for (int off = 16; off > 0; off >>= 1) s += __shfl_xor(s, off, 32);
    const float inv = 1.f / s;
    float2* dst = (float2*)(attn + ((size_t)bh * LL + qrow) * LL);
    for (int j = lane; j < 515; j += 32) {
      float2 e = Sr[j];
      dst[j] = make_float2(e.x * inv, e.y * inv);
    }
  }
}

// ---------------------------------------------------------------------------
// 5) out = P @ V per (bh, q-tile); 4 waves each own one 16-wide dv slice.
// ---------------------------------------------------------------------------
__global__ __launch_bounds__(128) void attnv_kernel(
    const float* __restrict__ attn, const float* __restrict__ vm,
    float* __restrict__ ao) {
  __shared__ float Ps[16][20];
  __shared__ float Vs[16][68];
  const int qt = blockIdx.x, bh = blockIdx.y;
  const int b = bh >> 4, h = bh & 15;
  const int tid = threadIdx.x, wid = tid >> 5, lane = tid & 31;
  const int lrow = lane & 15, kb = (lane >> 4) * 2;
  const float* arow = attn + (size_t)bh * LL * LL;
  const float* vb = vm + (size_t)b * LL * DD + h * DKV;
  // P tile: thread -> row tid>>3, 2 cols (tid&7)*2 (attn rows are 8B-aligned)
  const int pr = tid >> 3, pc = (tid & 7) * 2;
  const int pqrow = qt * 16 + pr;
  v8f acc = {0.f, 0.f, 0.f, 0.f, 0.f, 0.f, 0.f, 0.f};

  for (int kt = 0; kt < LT; ++kt) {
    {
      const int key = kt * 16 + pc;     // pair never straddles the L boundary
      float2 pv = make_float2(0.f, 0.f);
      if (pqrow < LL && key < LL)
        pv = *(const float2*)(arow + (size_t)pqrow * LL + key);
      *(float2*)&Ps[pr][pc] = pv;
    }
#pragma unroll
    for (int i = 0; i < 2; ++i) {       // V tile 16x64 via b128
      int idx = tid + i * 128;
      int r = idx >> 4, c4 = (idx & 15) * 4;
      int key = kt * 16 + r;
      v4f vv = {0.f, 0.f, 0.f, 0.f};
      if (key < LL) vv = *(const v4f*)(vb + (size_t)key * DD + c4);
      *(v4f*)&Vs[r][c4] = vv;
    }
    __syncthreads();
#pragma unroll
    for (int ks = 0; ks < 16; ks += 4) {
      v2f a, bf;
      a.x = Ps[lrow][ks + kb];
      a.y = Ps[lrow][ks + kb + 1];
      bf.x = Vs[ks + kb][wid * 16 + lrow];
      bf.y = Vs[ks + kb + 1][wid * 16 + lrow];
      acc = wmma_f32(a, bf, acc);
    }
    __syncthreads();
  }
  const int rh = (lane >> 4) * 8;
#pragma unroll
  for (int i = 0; i < 8; ++i) {
    int qrow = qt * 16 + i + rh;
    if (qrow < LL)
      ao[((size_t)b * LL + qrow) * DD + h * DKV + wid * 16 + lrow] = acc[i];
  }
}

// ---------------------------------------------------------------------------
extern "C" void kernel_launch(void* const* d_in, const int* in_sizes, int n_in,
                              void* d_out, int out_size, void* d_ws, size_t ws_size,
                              hipStream_t stream) {
  const float* x    = (const float*)d_in[0];
  const float* wq0  = (const float*)d_in[1];
  const float* wq1  = (const float*)d_in[2];
  const float* wk0  = (const float*)d_in[3];
  const float* wk1  = (const float*)d_in[4];
  const float* wv0  = (const float*)d_in[5];
  const float* wv1  = (const float*)d_in[6];
  const float* fc_w = (const float*)d_in[7];
  const float* fc_b = (const float*)d_in[8];
  const float* ln_g = (const float*)d_in[9];
  const float* ln_b = (const float*)d_in[10];

  float* out  = (float*)d_out;                        // (B, L, D)
  float* attn = out + (size_t)BB * LL * DD;           // (B, H, L, L)

  float* ws = (float*)d_ws;                           // needs ~169 MB
  const size_t plane = (size_t)NTOK * DD;
  float* xn = ws;
  float* q  = xn + plane;
  float* k  = q + plane;
  float* v  = k + plane;
  float* ao = v + plane;

  // 1) LayerNorm
  ln_kernel<<<NTOK, 256, 0, stream>>>(x, ln_g, ln_b, xn);

  // 2) Q/K/V projections with weight set 0 (fixup overwrites rep tokens)
  dim3 ggrid(DD / 64, (NTOK + 63) / 64);
  gemm64_kernel<<<ggrid, 256, 0, stream>>>(xn, wq0, q, NTOK, DD, DD, nullptr, nullptr);
  gemm64_kernel<<<ggrid, 256, 0, stream>>>(xn, wk0, k, NTOK, DD, DD, nullptr, nullptr);
  gemm64_kernel<<<ggrid, 256, 0, stream>>>(xn, wv0, v, NTOK, DD, DD, nullptr, nullptr);

  // 3) rep tokens (l >= 1027) use weight set 1
  fixup_kernel<<<dim3(BB * 3, 3), 256, 0, stream>>>(xn, wq1, wk1, wv1, q, k, v);

  // 4) fused scores + softmax -> attn region of d_out
  const size_t smem = (16 * 1040 + 16 * 68) * sizeof(float);   // ~69 KB LDS
  scores_softmax_kernel<<<dim3(LT, BB * HH), 256, smem, stream>>>(q, k, attn);

  // 5) P @ V -> (B, L, H*DV)
  attnv_kernel<<<dim3(LT, BB * HH), 128, 0, stream>>>(attn, v, ao);

  // 6) FC + bias + residual -> out region of d_out
  gemm64_kernel<<<ggrid, 256, 0, stream>>>(ao, fc_w, out, NTOK, DD, DD, fc_b, x);
}